// NUTS3_89017492177204
// MI455X (gfx1250) — compile-verified
//
#include <hip/hip_runtime.h>
#include <cstdint>

// ---------------------------------------------------------------------------
// NUTS (fixed-shape, jit-style full trees) on a diagonal Gaussian target.
// One workgroup == one chain; theta/r/prec slices live in VGPRs (float4),
// trajectory endpoints + binary-counter stack live in a private per-chain
// slice of d_ws (~136 MiB total -> resident in MI455X's 192 MB L2).
// Reductions: V_WMMA_F32_16X16X4_F32 ones-matrix trick + ds_swizzle SWAPX16.
// Stack copies: GLOBAL_{LOAD,STORE}_ASYNC_*_LDS_B128 (ASYNCcnt) when available.
// ---------------------------------------------------------------------------

namespace {

constexpr int   kB        = 128;
constexpr int   kD        = 16384;
constexpr int   kThreads  = 1024;             // 32 waves (wave32)
constexpr int   kEpt      = kD / kThreads;    // 16 floats per thread
constexpr int   kChunk    = kEpt / 4;         // 4 float4 per thread
constexpr int   kMaxDepth = 5;
constexpr float kEps      = 0.05f;
constexpr float kDeltaMax = 1000.0f;

// Per-chain workspace array slots (each kD floats).
enum {
  A_TH_R = 0, A_R_R, A_TH_F, A_R_F,
  A_NEAR_TH0 = 4,   // stack levels 0..3: near theta
  A_NEAR_R0  = 8,   // stack levels 0..3: near momentum
  A_PROP0    = 12,  // stack levels 0..3: subtree proposal
  A_PROP_CUR = 16,  // current subtree proposal
  kArrays    = 17
};
// ws requirement: 128 chains * 17 * 16384 * 4B = 136 MiB.

typedef __attribute__((ext_vector_type(2))) float v2f;
typedef __attribute__((ext_vector_type(4))) float v4f;
typedef __attribute__((ext_vector_type(8))) float v8f;
typedef __attribute__((ext_vector_type(4))) int   v4i;

#if __has_builtin(__builtin_amdgcn_global_load_async_to_lds_b128) && \
    __has_builtin(__builtin_amdgcn_global_store_async_from_lds_b128) && \
    __has_builtin(__builtin_amdgcn_s_wait_asynccnt)
#define NUTS_ASYNC_LDS 1
typedef __attribute__((address_space(1))) v4i gv4i;   // global (AS1) int4
typedef __attribute__((address_space(3))) v4i lv4i;   // LDS    (AS3) int4
#else
#define NUTS_ASYNC_LDS 0
#endif

// ---------------- counter-based RNG (stateless, deterministic) -------------
__device__ __forceinline__ uint32_t mix64(uint64_t x) {
  x ^= x >> 33; x *= 0xff51afd7ed558ccdULL;
  x ^= x >> 33; x *= 0xc4ceb9fe1a85ec53ULL;
  x ^= x >> 33;
  return (uint32_t)x;
}
__device__ __forceinline__ uint64_t tagrng(uint32_t stream, uint32_t a, uint32_t b) {
  return ((uint64_t)stream << 56) ^ ((uint64_t)a << 32) ^ (uint64_t)b;
}
__device__ __forceinline__ float u01(uint64_t x) {
  return (float)(mix64(x) >> 8) * (1.0f / 16777216.0f) + (0.5f / 16777216.0f);
}
__device__ __forceinline__ float nrnd(uint64_t x) {
  const float u1 = u01(x);
  const float u2 = u01(x ^ 0x9E3779B97F4A7C15ULL);
  return sqrtf(-2.0f * logf(u1)) * cosf(6.28318530717958647692f * u2);
}

// ---------------- wave/block reductions ------------------------------------
// Cross-half (lanes 0-15 <-> 16-31) add: ds_swizzle SWAPX16, full EXEC.
__device__ __forceinline__ float halfSwapAdd(float s) {
#if __has_builtin(__builtin_amdgcn_ds_swizzle)
  s += __int_as_float(__builtin_amdgcn_ds_swizzle(__float_as_int(s), 0x401f));
#else
  s += __shfl_xor(s, 16, 32);
#endif
  return s;
}

// Wave sum via V_WMMA_F32_16X16X4_F32 with an all-ones B: D rows become
// K-sums of the A tile; 8 adds + one half-swap finish the 32-lane sum.
__device__ __forceinline__ float waveSum(float p0, float p1) {
#if __has_builtin(__builtin_amdgcn_wmma_f32_16x16x4_f32)
  v2f a; a[0] = p0;   a[1] = p1;
  v2f b; b[0] = 1.0f; b[1] = 1.0f;
  v8f c = {};
  c = __builtin_amdgcn_wmma_f32_16x16x4_f32(false, a, false, b,
                                            (short)0, c, false, false);
  float s = ((c[0] + c[1]) + (c[2] + c[3])) + ((c[4] + c[5]) + (c[6] + c[7]));
  return halfSwapAdd(s);
#else
  float s = p0 + p1;
  for (int m = 16; m >= 1; m >>= 1) s += __shfl_xor(s, m, 32);
  return s;
#endif
}

__device__ __forceinline__ float blockSum(float v, float* red) {
  const float w  = waveSum(v, 0.0f);
  const int wave = threadIdx.x >> 5;
  const int lane = threadIdx.x & 31;
  if (lane == 0) red[wave] = w;
  __syncthreads();
  if (wave == 0) {                 // wave-uniform branch: EXEC stays all-1s
    float x = red[lane];           // exactly 32 wave partials
    x = waveSum(x, 0.0f);
    if (lane == 0) red[0] = x;
  }
  __syncthreads();
  const float s = red[0];
  __syncthreads();
  return s;
}

// Two independent sums sharing one LDS round-trip / barrier pair.
__device__ __forceinline__ void blockSum2(float a, float b, float* red,
                                          float& oa, float& ob) {
  const float wa = waveSum(a, 0.0f);
  const float wb = waveSum(b, 0.0f);
  const int wave = threadIdx.x >> 5;
  const int lane = threadIdx.x & 31;
  if (lane == 0) { red[wave] = wa; red[32 + wave] = wb; }
  __syncthreads();
  if (wave == 0) {
    float xa = red[lane];
    float xb = red[32 + lane];
    xa = waveSum(xa, 0.0f);
    xb = waveSum(xb, 0.0f);
    if (lane == 0) { red[0] = xa; red[32] = xb; }
  }
  __syncthreads();
  oa = red[0]; ob = red[32];
  __syncthreads();
}

// ---------------- array copy (global->LDS->global, async B128) -------------
__device__ __forceinline__ void copyArr(float* __restrict__ dst,
                                        const float* __restrict__ src,
                                        int t, v4f* stage) {
#if NUTS_ASYNC_LDS
#pragma unroll
  for (int c = 0; c < kChunk; ++c) {
    const int iv = t + c * kThreads;
    __builtin_amdgcn_global_load_async_to_lds_b128(
        (gv4i*)((const v4f*)src + iv), (lv4i*)(stage + iv), 0, 0);
  }
  __builtin_amdgcn_s_wait_asynccnt(0);
#pragma unroll
  for (int c = 0; c < kChunk; ++c) {
    const int iv = t + c * kThreads;
    __builtin_amdgcn_global_store_async_from_lds_b128(
        (gv4i*)((v4f*)dst + iv), (lv4i*)(stage + iv), 0, 0);
  }
  __builtin_amdgcn_s_wait_asynccnt(0);
#else
  (void)stage;
#pragma unroll
  for (int c = 0; c < kChunk; ++c) {
    const int iv = t + c * kThreads;
    ((v4f*)dst)[iv] = ((const v4f*)src)[iv];
  }
#endif
}

__global__ void __launch_bounds__(kThreads)
nuts_kernel(const float* __restrict__ theta0,
            const float* __restrict__ prec,
            float* __restrict__ out,
            float* __restrict__ ws) {
  __shared__ float red[64];
#if NUTS_ASYNC_LDS
  __shared__ v4f stage_s[kThreads * kChunk];   // 64 KiB staging
  v4f* stage = stage_s;
#else
  v4f* stage = nullptr;
#endif
  const int chain = blockIdx.x;
  const int t     = threadIdx.x;
  float* base = ws + (size_t)chain * ((size_t)kArrays * kD);
  const float* th0p = theta0 + (size_t)chain * kD;
  float* outp = out + (size_t)chain * kD;

  // Register-resident per-thread slice (float4 granularity).
  v4f thv[kChunk], rv[kChunk], prv[kChunk];
  float acc0 = 0.0f;
#pragma unroll
  for (int c = 0; c < kChunk; ++c) {
    const int iv = t + c * kThreads;
    thv[c] = ((const v4f*)th0p)[iv];
    prv[c] = ((const v4f*)prec)[iv];
#pragma unroll
    for (int q = 0; q < 4; ++q) {
      const uint32_t d = (uint32_t)(4 * iv + q);
      rv[c][q] = nrnd(tagrng(0, (uint32_t)chain, d));      // r0 ~ N(0,1)
      acc0 = fmaf(prv[c][q] * thv[c][q], thv[c][q], acc0); // prec*th^2
      acc0 = fmaf(rv[c][q], rv[c][q], acc0);               // r^2
    }
  }
#pragma unroll
  for (int c = 0; c < kChunk; ++c) {   // endpoints and theta_m init
    const int iv = t + c * kThreads;
    ((v4f*)(base + A_TH_R * kD))[iv] = thv[c];
    ((v4f*)(base + A_TH_F * kD))[iv] = thv[c];
    ((v4f*)(base + A_R_R  * kD))[iv] = rv[c];
    ((v4f*)(base + A_R_F  * kD))[iv] = rv[c];
    ((v4f*)outp)[iv] = thv[c];
  }
  // H0 = 0.5*(sum prec*th^2 + sum r^2): single fused reduction.
  const float H0   = 0.5f * blockSum(acc0, red);
  const float logu = logf(u01(tagrng(1, (uint32_t)chain, 0))) - H0;

  int n_tot = 1, s_tot = 1;
  uint32_t mc = 0;                 // merge RNG counter (uniform across chains)
  uint32_t nstP = 0, sstP = 0;     // bit-packed per-level stack scalars

  for (int j = 0; j < kMaxDepth; ++j) {
    const float vdir = (nrnd(tagrng(2, (uint32_t)j, 0)) >= 0.0f) ? 1.0f : -1.0f;
    const bool  neg  = (vdir < 0.0f);
    const float eps  = vdir * kEps;

    // Start from the chosen trajectory endpoint.
    const float* sth = base + (neg ? A_TH_R : A_TH_F) * kD;
    const float* sr  = base + (neg ? A_R_R  : A_R_F ) * kD;
#pragma unroll
    for (int c = 0; c < kChunk; ++c) {
      const int iv = t + c * kThreads;
      thv[c] = ((const v4f*)sth)[iv];
      rv[c]  = ((const v4f*)sr)[iv];
    }

    int n_c = 0, s_c = 0;
    const int leaves = 1 << j;
    float* const pcur = base + A_PROP_CUR * kD;

    for (int i = 0; i < leaves; ++i) {
      // ---- leapfrog (gradU = prec*theta), fused H partial ----
      float lh = 0.0f;
#pragma unroll
      for (int c = 0; c < kChunk; ++c) {
        rv[c]  += (-0.5f * eps) * prv[c] * thv[c];
        thv[c] += eps * rv[c];
        rv[c]  += (-0.5f * eps) * prv[c] * thv[c];
#pragma unroll
        for (int q = 0; q < 4; ++q) {
          lh = fmaf(prv[c][q] * thv[c][q], thv[c][q], lh);
          lh = fmaf(rv[c][q], rv[c][q], lh);
        }
      }
      const float H = 0.5f * blockSum(lh, red);   // e(theta) + 0.5*r.r
      n_c = (logu <= -H) ? 1 : 0;
      s_c = (logu - kDeltaMax < -H) ? 1 : 0;

      // Leaf proposal = current theta.
#pragma unroll
      for (int c = 0; c < kChunk; ++c)
        ((v4f*)pcur)[t + c * kThreads] = thv[c];

      // ---- merge completed subtrees (binary counter == recursion order) ----
      int ii = i, lvl = 0;
      while (ii & 1) {
        const int n_l = (int)((nstP >> (8 * lvl)) & 0xffu);
        const int s_l = (int)((sstP >> lvl) & 1u);
        const float rsel = u01(tagrng(3, (uint32_t)chain, mc)); ++mc;
        const bool take_right =
            (rsel * fmaxf((float)(n_l + n_c), 1.0f) < (float)n_c) && (s_l >= 1);
        if (!take_right)   // keep left subtree's proposal
          copyArr(pcur, base + (A_PROP0 + lvl) * kD, t, stage);

        // U-turn check: d = th_far - th_near; dots vs r_near and r_far.
        const float* nth = base + (A_NEAR_TH0 + lvl) * kD;
        const float* nr  = base + (A_NEAR_R0  + lvl) * kD;
        __builtin_prefetch(&nth[4 * t], 0, 0);
        __builtin_prefetch(&nr[4 * t], 0, 0);
        float p1 = 0.0f, p2 = 0.0f;
#pragma unroll
        for (int c = 0; c < kChunk; ++c) {
          const int iv = t + c * kThreads;
          const v4f a = ((const v4f*)nth)[iv];
          const v4f b = ((const v4f*)nr)[iv];
#pragma unroll
          for (int q = 0; q < 4; ++q) {
            const float dd = thv[c][q] - a[q];
            p1 = fmaf(dd, b[q], p1);
            p2 = fmaf(dd, rv[c][q], p2);
          }
        }
        float d1, d2;
        blockSum2(p1, p2, red, d1, d2);
        s_c = s_l * s_c * ((vdir * d1 >= 0.0f) ? 1 : 0)
                        * ((vdir * d2 >= 0.0f) ? 1 : 0);
        n_c = n_l + ((s_l >= 1) ? n_c : 0);
        ii >>= 1; ++lvl;
      }

      if (i != leaves - 1) {   // push completed subtree at level `lvl`
        if (lvl == 0) {        // near = current (leaf) state, from registers
          float* dth = base + A_NEAR_TH0 * kD;
          float* dr  = base + A_NEAR_R0  * kD;
#pragma unroll
          for (int c = 0; c < kChunk; ++c) {
            const int iv = t + c * kThreads;
            ((v4f*)dth)[iv] = thv[c];
            ((v4f*)dr)[iv]  = rv[c];
          }
        } else {               // near = near of last merged level (lvl-1)
          copyArr(base + (A_NEAR_TH0 + lvl) * kD,
                  base + (A_NEAR_TH0 + lvl - 1) * kD, t, stage);
          copyArr(base + (A_NEAR_R0 + lvl) * kD,
                  base + (A_NEAR_R0 + lvl - 1) * kD, t, stage);
        }
        copyArr(base + (A_PROP0 + lvl) * kD, pcur, t, stage);
        nstP = (nstP & ~(0xffu << (8 * lvl))) | ((uint32_t)n_c << (8 * lvl));
        sstP = (sstP & ~(1u   << lvl))        | ((uint32_t)s_c << lvl);
      }
    }

    // ---- Metropolis-within-NUTS accept of the subtree proposal ----
    const float ar = u01(tagrng(4, (uint32_t)chain, (uint32_t)j));
    const bool acc = (ar * (float)n_tot < (float)n_c) && (s_tot >= 1) && (s_c >= 1);
    if (acc) copyArr(outp, pcur, t, stage);
    n_tot += (s_tot >= 1) ? n_c : 0;

    // New far endpoint = final integrated state.
    float* eth = base + (neg ? A_TH_R : A_TH_F) * kD;
    float* er  = base + (neg ? A_R_R  : A_R_F ) * kD;
#pragma unroll
    for (int c = 0; c < kChunk; ++c) {
      const int iv = t + c * kThreads;
      ((v4f*)eth)[iv] = thv[c];
      ((v4f*)er)[iv]  = rv[c];
    }

    // Trajectory-level U-turn test.
    const float* thr = base + A_TH_R * kD;
    const float* thf = base + A_TH_F * kD;
    const float* rrr = base + A_R_R  * kD;
    const float* rrf = base + A_R_F  * kD;
    __builtin_prefetch(&thr[4 * t], 0, 0);
    float p1 = 0.0f, p2 = 0.0f;
#pragma unroll
    for (int c = 0; c < kChunk; ++c) {
      const int iv = t + c * kThreads;
      const v4f a  = ((const v4f*)thf)[iv];
      const v4f b  = ((const v4f*)thr)[iv];
      const v4f x  = ((const v4f*)rrr)[iv];
      const v4f y  = ((const v4f*)rrf)[iv];
#pragma unroll
      for (int q = 0; q < 4; ++q) {
        const float dd = a[q] - b[q];
        p1 = fmaf(dd, x[q], p1);
        p2 = fmaf(dd, y[q], p2);
      }
    }
    float d1, d2;
    blockSum2(p1, p2, red, d1, d2);
    s_tot = s_tot * s_c * ((d1 >= 0.0f) ? 1 : 0) * ((d2 >= 0.0f) ? 1 : 0);
  }
}

}  // namespace

extern "C" void kernel_launch(void* const* d_in, const int* in_sizes, int n_in,
                              void* d_out, int out_size, void* d_ws, size_t ws_size,
                              hipStream_t stream) {
  (void)in_sizes; (void)n_in; (void)out_size; (void)ws_size;
  const float* theta0 = (const float*)d_in[0];   // [B, D] f32
  const float* prec   = (const float*)d_in[1];   // [D] f32
  float* out = (float*)d_out;                    // [B, D] f32 (theta_m)
  float* ws  = (float*)d_ws;                     // needs >= 136 MiB
  nuts_kernel<<<kB, kThreads, 0, stream>>>(theta0, prec, out, ws);
}